// HebbianConv2d_81801947119726
// MI455X (gfx1250) — compile-verified
//
#include <hip/hip_runtime.h>
#include <hip/hip_bf16.h>

typedef __attribute__((ext_vector_type(2))) float v2f;
typedef __attribute__((ext_vector_type(8))) float v8f;
typedef __attribute__((ext_vector_type(4))) unsigned int u32x4;
typedef __attribute__((ext_vector_type(8))) int i32x8;
typedef __attribute__((ext_vector_type(4))) int i32x4;

// Problem constants
#define BB   32
#define CC   64
#define HH   64
#define WW   64
#define OO   128
#define KHW  3
#define HPW  62                 // 64 - 3 + 1
#define LL   (HPW * HPW)        // 3844 positions per image
#define NTOT (BB * LL)          // 123008
#define CK   (CC * KHW * KHW)   // 576
#define CHW  (CC * HH * WW)     // 262144
#define HW   (HH * WW)          // 4096
#define OL   (OO * LL)          // 492032
#define YTOT (BB * OO * LL)     // 15745024
#define NTILES (NTOT / 16)      // 7688

// Workspace layout (floats)
#define WS_WN     0                         // 73728 : normalized weights (O, CK)
#define WS_STAT   (WS_WN + OO * CK)         // 2*123008 : per-position (max, 1/denom)
#define WS_S1     (WS_STAT + 2 * NTOT)      // 128 : sum r per o
#define WS_S2     (WS_S1 + OO)              // 128 : sum r^2 per o
#define WS_DACC   (WS_S2 + OO)              // 73728 : delta accumulator (O, CK)

#if __has_builtin(__builtin_amdgcn_tensor_load_to_lds) && \
    __has_builtin(__builtin_amdgcn_s_wait_tensorcnt)
#define USE_TDM 1
#else
#define USE_TDM 0
#endif

__device__ __forceinline__ v8f wmma_f32(v2f a, v2f b, v8f c) {
    return __builtin_amdgcn_wmma_f32_16x16x4_f32(false, a, false, b, (short)0,
                                                 c, false, false);
}

#if USE_TDM
// 2D TDM load: tile_d0 x tile_d1 elements (4B), row stride stride_d0 elements,
// from gaddr into LDS byte address lds_byte.  Wave-uniform arguments.
__device__ __forceinline__ void tdm_load_2d(unsigned lds_byte, const void* gaddr,
                                            unsigned tensor_d0, unsigned tensor_d1,
                                            unsigned tile_d0, unsigned tile_d1,
                                            unsigned stride_d0) {
    unsigned long long ga = (unsigned long long)(uintptr_t)gaddr;
    u32x4 g0;
    g0.x = 1u;                                    // count=1, user descriptor
    g0.y = lds_byte;                              // LDS byte address
    g0.z = (unsigned)(ga & 0xFFFFFFFFu);          // global addr [31:0]
    g0.w = (unsigned)((ga >> 32) & 0x01FFFFFFu)   // global addr [56:32]
           | (2u << 30);                          // type = 2 ("image")
    i32x8 g1;
    g1[0] = (2 << 16);                            // data_size=4B, mask=0, no pad
    g1[1] = (int)((tensor_d0 & 0xFFFFu) << 16);   // tensor_dim0[15:0] @ bits63:48
    g1[2] = (int)(((tensor_d0 >> 16) & 0xFFFFu) |
                  ((tensor_d1 & 0xFFFFu) << 16)); // dim0[31:16] | dim1[15:0]
    g1[3] = (int)(((tensor_d1 >> 16) & 0xFFFFu) |
                  ((tile_d0 & 0xFFFFu) << 16));   // dim1[31:16] | tile_dim0
    g1[4] = (int)(tile_d1 & 0xFFFFu);             // tile_dim1, tile_dim2=0
    g1[5] = (int)stride_d0;                       // tensor_dim0_stride[31:0]
    g1[6] = 0;
    g1[7] = 0;
    i32x4 z4 = {0, 0, 0, 0};
    i32x8 z8 = {0, 0, 0, 0, 0, 0, 0, 0};
    __builtin_amdgcn_tensor_load_to_lds(g0, g1, z4, z4, z8, 0);
}
#endif

// ---------------------------------------------------------------------------
// Kernel 0: per-filter L2 norm -> wn; zero delta accumulator
// ---------------------------------------------------------------------------
__global__ void k_prep(const float* __restrict__ weight, float* __restrict__ ws) {
    __shared__ float red[64];
    const int o = blockIdx.x;
    const int tid = threadIdx.x;
    float* wn = ws + WS_WN;
    float* dacc = ws + WS_DACC;

    float ss = 0.f;
    for (int i = tid; i < CK; i += 64) {
        float w = weight[o * CK + i];
        ss += w * w;
    }
    red[tid] = ss;
    __syncthreads();
    for (int s = 32; s > 0; s >>= 1) {
        if (tid < s) red[tid] += red[tid + s];
        __syncthreads();
    }
    __shared__ float s_inv;
    if (tid == 0) {
        float nrm = sqrtf(red[0]);
        s_inv = (nrm == 0.f) ? 1.f : (1.f / nrm);
    }
    __syncthreads();
    const float inv = s_inv;
    for (int i = tid; i < CK; i += 64) {
        wn[o * CK + i] = weight[o * CK + i] * inv;
        dacc[o * CK + i] = 0.f;
    }
}

// ---------------------------------------------------------------------------
// Kernel 1: conv (implicit GEMM, WMMA f32 16x16x4) + bias + fused softmax stats
// A-fragments staged through LDS by the Tensor Data Mover (double buffered).
// grid: 961 blocks x 256 threads; 1 wave = one 16-position tile, all 128 O.
// ---------------------------------------------------------------------------
__global__ void __launch_bounds__(256)
k_conv(const float* __restrict__ x, const float* __restrict__ bias,
       const float* __restrict__ ws_ro, float* __restrict__ ws,
       float* __restrict__ yout) {
    __shared__ int offTab[CK];
    __shared__ float sbias[OO];
#if USE_TDM
    __shared__ float wnLds[2][OO * 16];      // two 8KB slabs: 128 rows x 16 k
#endif
    for (int i = threadIdx.x; i < CK; i += 256) {
        int c = i / 9, r = i % 9;
        offTab[i] = c * HW + (r / 3) * WW + (r % 3);
    }
    if (threadIdx.x < OO) sbias[threadIdx.x] = bias[threadIdx.x];

    const float* wn = ws_ro + WS_WN;
    v2f* statArr = (v2f*)(ws + WS_STAT);

    const int lane = threadIdx.x & 31;
    const int wid = threadIdx.x >> 5;
    const int nBase = (blockIdx.x * 8 + wid) * 16;
    const int m = lane & 15;      // A: row M / B: col N
    const int hi = lane >> 4;     // lane half selects K pair
    const int koff = hi * 2;

    // this lane's position (column of the conv GEMM)
    const int nl = nBase + m;
    const int b = nl / LL, l = nl % LL;
    const int h0 = l / HPW, w0 = l % HPW;
    const float* px = x + (size_t)b * CHW + h0 * WW + w0;

    v8f acc[8] = {};

#if USE_TDM
    // prime slab 0
    if (wid == 0) {
        tdm_load_2d((unsigned)(uintptr_t)&wnLds[0][0], wn, CK, OO, 16, OO, CK);
        __builtin_amdgcn_s_wait_tensorcnt(0);
    }
    __syncthreads();

    int buf = 0;
    for (int s = 0; s < CK / 16; ++s) {          // 36 k-slabs of 16
        if (wid == 0 && s + 1 < CK / 16)
            tdm_load_2d((unsigned)(uintptr_t)&wnLds[buf ^ 1][0],
                        wn + (s + 1) * 16, CK, OO, 16, OO, CK);
        const int kbase = s * 16;
        const float* slab = &wnLds[buf][0];
#pragma unroll
        for (int ks = 0; ks < 4; ++ks) {
            const int kk = kbase + ks * 4 + koff;
            const int off0 = offTab[kk];
            const int off1 = offTab[kk + 1];
            v2f bf;
            bf.x = px[off0];
            bf.y = px[off1];
            const int lk = ks * 4 + koff;
#pragma unroll
            for (int t = 0; t < 8; ++t) {
                const v2f af = *(const v2f*)(slab + (t * 16 + m) * 16 + lk);
                acc[t] = wmma_f32(af, bf, acc[t]);
            }
        }
        if (wid == 0 && s + 1 < CK / 16) __builtin_amdgcn_s_wait_tensorcnt(0);
        __syncthreads();
        buf ^= 1;
    }
#else
    __syncthreads();
    for (int k = 0; k < CK; k += 4) {
        const int kk = k + koff;
        const int off0 = offTab[kk];
        const int off1 = offTab[kk + 1];
        v2f bf;
        bf.x = px[off0];
        bf.y = px[off1];
#pragma unroll
        for (int t = 0; t < 8; ++t) {
            const v2f af = *(const v2f*)(wn + (size_t)(t * 16 + m) * CK + kk);
            acc[t] = wmma_f32(af, bf, acc[t]);
        }
    }
#endif

    // epilogue: bias, store y, softmax stats for the 16 positions of this tile
#pragma unroll
    for (int t = 0; t < 8; ++t)
#pragma unroll
        for (int v = 0; v < 8; ++v)
            acc[t][v] += sbias[t * 16 + v + 8 * hi];

    const size_t ybase = (size_t)b * OL + (size_t)l;
#pragma unroll
    for (int t = 0; t < 8; ++t)
#pragma unroll
        for (int v = 0; v < 8; ++v) {
            const int o = t * 16 + v + 8 * hi;
            yout[ybase + (size_t)o * LL] = acc[t][v];
        }

    // lane pair (lane, lane^16) holds the full 128-channel column for position nl
    float mx = -3.402823466e+38f;
#pragma unroll
    for (int t = 0; t < 8; ++t)
#pragma unroll
        for (int v = 0; v < 8; ++v) mx = fmaxf(mx, acc[t][v]);
    mx = fmaxf(mx, __shfl_xor(mx, 16, 32));

    float se = 0.f;
#pragma unroll
    for (int t = 0; t < 8; ++t)
#pragma unroll
        for (int v = 0; v < 8; ++v) se += __expf(acc[t][v] - mx);
    se += __shfl_xor(se, 16, 32);

    if (hi == 0) {
        v2f st;
        st.x = mx;
        st.y = 1.f / se;
        statArr[nl] = st;
    }
}

// ---------------------------------------------------------------------------
// Kernel 2: per-o sums s1 = sum r, s2 = sum r^2
// ---------------------------------------------------------------------------
__global__ void __launch_bounds__(256)
k_rsum(const float* __restrict__ y, const float* __restrict__ ws_ro,
       float* __restrict__ ws) {
    __shared__ float r1[256], r2[256];
    const int o = blockIdx.x;
    const int tid = threadIdx.x;
    const v2f* statArr = (const v2f*)(ws_ro + WS_STAT);

    float a = 0.f, c = 0.f;
    for (int n = tid; n < NTOT; n += 256) {
        const int b = n / LL, l = n % LL;
        const float yv = y[(size_t)b * OL + (size_t)o * LL + l];
        const v2f st = statArr[n];
        const float r = __expf(yv - st.x) * st.y;
        a += r;
        c += r * r;
    }
    r1[tid] = a;
    r2[tid] = c;
    __syncthreads();
    for (int s = 128; s > 0; s >>= 1) {
        if (tid < s) { r1[tid] += r1[tid + s]; r2[tid] += r2[tid + s]; }
        __syncthreads();
    }
    if (tid == 0) {
        ws[WS_S1 + o] = r1[0];
        ws[WS_S2 + o] = r2[0];
    }
}

// ---------------------------------------------------------------------------
// Kernel 3: delta GEMM  acc[o][ck] += sum_n r^2[o][n] * patch[n][ck]
// WMMA f32 16x16x4: M=o, N=ck, K=positions (split-K via grid-stride + atomics)
// ---------------------------------------------------------------------------
__global__ void __launch_bounds__(256)
k_delta(const float* __restrict__ x, const float* __restrict__ y,
        const float* __restrict__ ws_ro, float* __restrict__ ws) {
    __shared__ int offTab[CK];
    for (int i = threadIdx.x; i < CK; i += 256) {
        int c = i / 9, r = i % 9;
        offTab[i] = c * HW + (r / 3) * WW + (r % 3);
    }
    __syncthreads();

    const v2f* statArr = (const v2f*)(ws_ro + WS_STAT);
    float* dacc = ws + WS_DACC;

    const int lane = threadIdx.x & 31;
    const int wid = threadIdx.x >> 5;
    const int oBase = wid * 16;
    const int m = lane & 15;   // A row (o%16) and B col (ck%16)
    const int hi = lane >> 4;
    const int koff = hi * 2;
    const int oRow = oBase + m;

    for (int g = 0; g < 9; ++g) {           // 9 groups of 4 ck-tiles (64 cols)
        int coff[4];
#pragma unroll
        for (int t = 0; t < 4; ++t) coff[t] = offTab[g * 64 + t * 16 + m];

        v8f acc[4] = {};
        for (int nt = blockIdx.x; nt < NTILES; nt += gridDim.x) {
            const int nb = nt * 16;
            // hint the next grid-stride tile's stats into cache
            __builtin_prefetch(statArr + nb + 16 * gridDim.x, 0, 1);
#pragma unroll
            for (int step = 0; step < 4; ++step) {
                const int n0 = nb + step * 4 + koff;
                const int n1 = n0 + 1;
                const int b0 = n0 / LL, l0 = n0 % LL;
                const int b1 = n1 / LL, l1 = n1 % LL;
                const int h00 = l0 / HPW, w00 = l0 % HPW;
                const int h01 = l1 / HPW, w01 = l1 % HPW;
                const int x0 = b0 * CHW + h00 * WW + w00;
                const int x1 = b1 * CHW + h01 * WW + w01;

                const float y0 = y[(size_t)b0 * OL + (size_t)oRow * LL + l0];
                const float y1 = y[(size_t)b1 * OL + (size_t)oRow * LL + l1];
                const v2f s0 = statArr[n0];
                const v2f s1 = statArr[n1];
                float r0 = __expf(y0 - s0.x) * s0.y; r0 *= r0;
                float r1 = __expf(y1 - s1.x) * s1.y; r1 *= r1;
                v2f af; af.x = r0; af.y = r1;
#pragma unroll
                for (int t = 0; t < 4; ++t) {
                    v2f bf;
                    bf.x = x[x0 + coff[t]];
                    bf.y = x[x1 + coff[t]];
                    acc[t] = wmma_f32(af, bf, acc[t]);
                }
            }
        }
#pragma unroll
        for (int t = 0; t < 4; ++t)
#pragma unroll
            for (int v = 0; v < 8; ++v) {
                const int o = oBase + v + 8 * hi;
                const int ck = g * 64 + t * 16 + m;
                atomicAdd(dacc + (size_t)o * CK + ck, acc[t][v]);
            }
    }
}

// ---------------------------------------------------------------------------
// Kernel 4: delta_w = dacc/s1 - (s2/s1)*weight
// ---------------------------------------------------------------------------
__global__ void k_final(const float* __restrict__ weight,
                        const float* __restrict__ ws_ro,
                        float* __restrict__ dout) {
    const int i = blockIdx.x * blockDim.x + threadIdx.x;
    if (i >= OO * CK) return;
    const int o = i / CK;
    float s1 = ws_ro[WS_S1 + o];
    const float s2 = ws_ro[WS_S2 + o];
    if (s1 == 0.f) s1 = 1.f;
    const float inv = 1.f / s1;
    dout[YTOT + i] = ws_ro[WS_DACC + i] * inv - (s2 * inv) * weight[i];
}

// ---------------------------------------------------------------------------
extern "C" void kernel_launch(void* const* d_in, const int* in_sizes, int n_in,
                              void* d_out, int out_size, void* d_ws, size_t ws_size,
                              hipStream_t stream) {
    const float* x = (const float*)d_in[0];
    const float* weight = (const float*)d_in[1];
    const float* bias = (const float*)d_in[2];
    float* out = (float*)d_out;
    float* ws = (float*)d_ws;

    k_prep<<<OO, 64, 0, stream>>>(weight, ws);
    k_conv<<<NTILES / 8, 256, 0, stream>>>(x, bias, ws, ws, out);
    k_rsum<<<OO, 256, 0, stream>>>(out, ws, ws);
    k_delta<<<160, 256, 0, stream>>>(x, out, ws, ws);
    k_final<<<(OO * CK + 255) / 256, 256, 0, stream>>>(weight, ws, out);
}